// QuantumClassifierModel_65481071409992
// MI455X (gfx1250) — compile-verified
//
#include <hip/hip_runtime.h>
#include <hip/hip_bf16.h>

#define NQ     14
#define DEPTH  6
#define DIM    (1 << NQ)          // 16384
#define THREADS 256               // 8 waves of 32

typedef __attribute__((ext_vector_type(2))) float v2f;
typedef __attribute__((ext_vector_type(8))) float v8f;

// LDS layout (floats)
#define OFF_RE    0
#define OFF_IM    (DIM)
#define OFF_CT    (2 * DIM)                    // cos(theta/2), NQ*DEPTH
#define OFF_ST    (OFF_CT + NQ * DEPTH)        // sin(theta/2), NQ*DEPTH
#define OFF_XC    (OFF_ST + NQ * DEPTH)        // cos(x/2), NQ
#define OFF_XS    (OFF_XC + NQ)                // sin(x/2), NQ
#define OFF_ZE    (OFF_XS + NQ)                // zexp, NQ
#define SMEM_FLOATS (OFF_ZE + NQ + 2)
#define SMEM_BYTES  (SMEM_FLOATS * sizeof(float))

__global__ __launch_bounds__(THREADS, 2)
void qsim_wmma_kernel(const float* __restrict__ x,
                      const float* __restrict__ theta,
                      const float* __restrict__ head_w,
                      const float* __restrict__ head_b,
                      float* __restrict__ out)
{
    extern __shared__ float smem[];
    float* sre = smem + OFF_RE;
    float* sim = smem + OFF_IM;
    float* sct = smem + OFF_CT;
    float* sst = smem + OFF_ST;
    float* sxc = smem + OFF_XC;
    float* sxs = smem + OFF_XS;
    float* sze = smem + OFF_ZE;

    const int tid   = threadIdx.x;
    const int b     = blockIdx.x;
    // Wave index is uniform across the wave: broadcast it into an SGPR so the
    // tile loop below compiles to scalar loop control (EXEC provably all-ones
    // around every WMMA, per ISA 7.12 requirement) instead of exec masking.
    const int wave  = __builtin_amdgcn_readfirstlane(tid >> 5);
    const int lane  = tid & 31;
    const int lhalf = lane >> 4;     // which half of the wave
    const int l15   = lane & 15;

    // ---- precompute rotation cos/sin and input-angle cos/sin ----
    if (tid < NQ * DEPTH) {
        float t = 0.5f * theta[tid];
        sct[tid] = __cosf(t);
        sst[tid] = __sinf(t);
    }
    if (tid < NQ) {
        float h = 0.5f * x[b * NQ + tid];
        sxc[tid] = __cosf(h);
        sxs[tid] = __sinf(h);
    }
    __syncthreads();

    // ---- initial product state: amp(i) = (prod mags) * (-i)^popcount(i) ----
    for (int i = tid; i < DIM; i += THREADS) {
        float mag = 1.0f;
        #pragma unroll
        for (int q = 0; q < NQ; ++q) {
            int bit = (i >> (NQ - 1 - q)) & 1;
            mag *= bit ? sxs[q] : sxc[q];
        }
        int k = __popc(i) & 3;   // (-i)^k: 1, -i, -1, +i
        float re = (k == 0) ? mag : ((k == 2) ? -mag : 0.0f);
        float im = (k == 1) ? -mag : ((k == 3) ? mag : 0.0f);
        sre[i] = re;
        sim[i] = im;
    }
    __syncthreads();

    // ---- 6 layers; each layer = 4 fused-4-qubit WMMA sweeps + CZ sign ----
    for (int d = 0; d < DEPTH; ++d) {
        #pragma unroll
        for (int g = 0; g < 4; ++g) {
            const int lowbits = (g == 0) ? 10 : (g == 1) ? 6 : (g == 2) ? 2 : 0;

            // Build fused 16x16 gate G4 = kron of four 2x2 RY's, in the
            // WMMA A layout: chunk j holds K=4j..4j+3;
            // lane (l15,lhalf), VGPR v  ->  A[m= l15][k = 4j + v + 2*lhalf]
            v2f A[4];
            #pragma unroll
            for (int j = 0; j < 4; ++j) {
                #pragma unroll
                for (int v = 0; v < 2; ++v) {
                    const int kk = 4 * j + v + 2 * lhalf;
                    const int mm = l15;
                    float val = 1.0f;
                    #pragma unroll
                    for (int t = 0; t < 4; ++t) {
                        const int mb = (mm >> (3 - t)) & 1;
                        const int kb = (kk >> (3 - t)) & 1;
                        float e;
                        if (g == 3 && t < 2) {
                            // identity padding on qubits 10,11 (done in g==2)
                            e = (mb == kb) ? 1.0f : 0.0f;
                        } else {
                            const int qq = (g < 3) ? (g * 4 + t) : (10 + t);
                            const float ct = sct[d * NQ + qq];
                            const float st = sst[d * NQ + qq];
                            e = (mb == 0) ? (kb == 0 ? ct : -st)
                                          : (kb == 0 ? st :  ct);
                        }
                        val *= e;
                    }
                    A[j][v] = val;
                }
            }

            // 1024 columns / 16-per-tile = 64 tiles, x2 (re,im) = 128 tasks.
            // Each wave owns whole tiles -> in-place transform is race-free.
            // `task` starts from an SGPR and has a uniform trip count, so this
            // is a scalar-controlled loop: no exec-mask churn around WMMA.
            for (int task = wave; task < 128; task += 8) {
                float* sp = (task & 1) ? sim : sre;
                const int ctile = (task >> 1) * 16;

                // B operand (state tile), mirrored striping of A:
                // lane column n = l15; VGPR v holds rows {4j+v, 4j+v+2}
                const int c    = ctile + l15;
                const int base = ((c >> lowbits) << (lowbits + 4)) |
                                 (c & ((1 << lowbits) - 1));
                v2f Bv[4];
                #pragma unroll
                for (int j = 0; j < 4; ++j) {
                    #pragma unroll
                    for (int v = 0; v < 2; ++v) {
                        const int kk = 4 * j + v + 2 * lhalf;
                        Bv[j][v] = sp[base + (kk << lowbits)];
                    }
                }

                v8f Cacc = {};
                #pragma unroll
                for (int j = 0; j < 4; ++j) {
                    Cacc = __builtin_amdgcn_wmma_f32_16x16x4_f32(
                        false, A[j], false, Bv[j], (short)0, Cacc, false, false);
                }

                // D layout: VGPR v -> row m = v + 8*lhalf, column n = l15.
                // Fuse the CZ-chain diagonal sign into the last sweep.
                #pragma unroll
                for (int v = 0; v < 8; ++v) {
                    const int m   = v + 8 * lhalf;
                    const int idx = base + (m << lowbits);
                    float val = Cacc[v];
                    if (g == 3) {
                        const int par = __popc(idx & (idx >> 1)) & 1;
                        val = par ? -val : val;
                    }
                    sp[idx] = val;
                }
            }
            __syncthreads();
        }
    }

    // ---- per-qubit <Z>: deterministic two-stage reduction (serial final sum) ----
    float acc[NQ];
    #pragma unroll
    for (int q = 0; q < NQ; ++q) acc[q] = 0.0f;

    for (int i = tid; i < DIM; i += THREADS) {
        const float re = sre[i], im = sim[i];
        const float p = re * re + im * im;
        #pragma unroll
        for (int q = 0; q < NQ; ++q) {
            acc[q] += ((i >> (NQ - 1 - q)) & 1) ? -p : p;
        }
    }
    __syncthreads();   // done reading state; reuse sre as scratch

    #pragma unroll
    for (int q = 0; q < NQ; ++q) sre[q * THREADS + tid] = acc[q];
    __syncthreads();

    if (tid < NQ) {
        float s = 0.0f;
        for (int t = 0; t < THREADS; ++t) s += sre[tid * THREADS + t];
        sze[tid] = s;
    }
    __syncthreads();

    // ---- head: logits[o] = b[o] + sum_i w[o,i] * zexp[NQ-1-i] ----
    if (tid < 2) {
        float s = head_b[tid];
        #pragma unroll
        for (int i = 0; i < NQ; ++i) {
            s += head_w[tid * NQ + i] * sze[NQ - 1 - i];
        }
        out[b * 2 + tid] = s;
    }
}

extern "C" void kernel_launch(void* const* d_in, const int* in_sizes, int n_in,
                              void* d_out, int out_size, void* d_ws, size_t ws_size,
                              hipStream_t stream)
{
    const float* x      = (const float*)d_in[0];
    const float* theta  = (const float*)d_in[1];
    const float* head_w = (const float*)d_in[2];
    const float* head_b = (const float*)d_in[3];
    float* out          = (float*)d_out;

    const int batch = in_sizes[0] / NQ;   // 512

    qsim_wmma_kernel<<<batch, THREADS, SMEM_BYTES, stream>>>(
        x, theta, head_w, head_b, out);
}